// FaceAlignmentModule_63522566307967
// MI455X (gfx1250) — compile-verified
//
#include <hip/hip_runtime.h>
#include <hip/hip_bf16.h>
#include <math.h>

typedef __attribute__((ext_vector_type(2))) float v2f;
typedef __attribute__((ext_vector_type(8))) float v8f;

#define N_IMG   16
#define NCH     98
#define HH      256
#define WW      256
#define PLANE   (HH*WW)
#define NPLANES (N_IMG*NCH)
#define NSLOT   48          // 48 slot-pairs = 96 virtual channel halves
#define NROUND  24          // 2 slots per WMMA round
#define XS_SPLIT_ROUND 10   // rounds 0..10 (slots 0..21) = the x_ subset

// ---------------- compile-time slot schedule --------------------------------
// _shift(v,n) = roll(v,-n,axis=H)  => out[h] = in[(h+n) mod 256]
// Each slot supplies one B element for both wave halves:
//   lanes 0-15 use (cA,shA,mA), lanes 16-31 use (cB,shB,mB); power p shared.
struct Slot { int cA, cB, shA, shB, p; float mA, mB; };
struct STab { Slot s[NSLOT]; };

constexpr STab make_slots() {
  STab t{};
  int i = 0;
  // ---- x_ subset: rounds 0..10 (43 real virtual channels + 1 dummy) ----
  t.s[i++] = {51, 96, 0, 0, 1, 1.0f, 1.0f};
  t.s[i++] = {97, 54, 0, 4, 1, 1.0f, 1.0f};   // cross-pair, both p=1
  t.s[i++] = {52, 53, 4, 4, 1, 1.0f, 1.0f};
  t.s[i++] = {55, 56, 8, 8, 4, 1.0f, 1.0f};
  t.s[i++] = {57, 58, 8, 8, 4, 1.0f, 1.0f};
  t.s[i++] = {59, 59, 8, 8, 4, 1.0f, 0.0f};   // dummy half B
  for (int c = 60; c <= 74; c += 2) t.s[i++] = {c, c + 1,  -8,  -8, 3, 1.0f, 1.0f}; // eyes ^3
  for (int c = 60; c <= 74; c += 2) t.s[i++] = {c, c + 1, -24, -24, 1, 1.0f, 1.0f}; // eyes ^1
  // ---- x-only channels: rounds 11..23 (51 real + 1 dummy) ----
  for (int c = 8;  c <= 22; c += 2) t.s[i++] = {c, c + 1, 0, 0, 3, 1.0f, 1.0f};
  t.s[i++] = {24, 24, 0, 0, 3, 1.0f, 0.0f};   // dummy half B
  for (int c = 34; c <= 44; c += 2) t.s[i++] = {c, c + 1, -7, -7, 2, 1.0f, 1.0f};
  t.s[i++] = {47, 48, -7, -7, 2, 1.0f, 1.0f};
  t.s[i++] = {49, 50, -7, -7, 2, 1.0f, 1.0f};
  t.s[i++] = {77, 78, -8, -8, 4, 1.0f, 1.0f};
  t.s[i++] = {79, 80, -8, -8, 4, 1.0f, 1.0f};
  t.s[i++] = {83, 84,  8,  8, 4, 1.0f, 1.0f};
  t.s[i++] = {85, 86,  8,  8, 4, 1.0f, 1.0f};
  t.s[i++] = {81, 87, -8,  8, 4, 1.0f, 1.0f}; // cross-pair, both p=4
  for (int c = 88; c <= 94; c += 2) t.s[i++] = {c, c + 1, -2, -2, 3, 1.0f, 1.0f};
  return t;
}
constexpr STab g_slots = make_slots();   // value-only (constexpr) use => immediates

// ---------------- Pass 1: per-plane argmax / min / max ----------------------
__global__ __launch_bounds__(256) void plane_reduce_kernel(
    const float4* __restrict__ hm4, int* __restrict__ idx_out,
    float* __restrict__ scale_out, float* __restrict__ mn_out) {
  const int plane = blockIdx.x;                 // 0..1567
  const float4* p = hm4 + (size_t)plane * (PLANE / 4);
  const int tid = threadIdx.x;

  float mx = -3.402823466e38f, mn = 3.402823466e38f;
  int mi = 0;
  #pragma unroll 4
  for (int i = tid; i < PLANE / 4; i += 256) {
    float4 v = p[i];
    int b = 4 * i;
    if (v.x > mx) { mx = v.x; mi = b + 0; }
    if (v.y > mx) { mx = v.y; mi = b + 1; }
    if (v.z > mx) { mx = v.z; mi = b + 2; }
    if (v.w > mx) { mx = v.w; mi = b + 3; }
    mn = fminf(mn, fminf(fminf(v.x, v.y), fminf(v.z, v.w)));
  }
  // wave32 reduction (argmax keeps lowest index on ties = jnp.argmax semantics)
  #pragma unroll
  for (int off = 16; off > 0; off >>= 1) {
    float omx = __shfl_down(mx, off, 32);
    int   omi = __shfl_down(mi, off, 32);
    float omn = __shfl_down(mn, off, 32);
    if (omx > mx || (omx == mx && omi < mi)) { mx = omx; mi = omi; }
    mn = fminf(mn, omn);
  }
  __shared__ float smx[8], smn[8];
  __shared__ int   smi[8];
  if ((tid & 31) == 0) { smx[tid >> 5] = mx; smi[tid >> 5] = mi; smn[tid >> 5] = mn; }
  __syncthreads();
  if (tid == 0) {
    for (int k = 1; k < 8; ++k) {
      if (smx[k] > mx || (smx[k] == mx && smi[k] < mi)) { mx = smx[k]; mi = smi[k]; }
      mn = fminf(mn, smn[k]);
    }
    float mxt = (mx >= 0.1f) ? mx : 0.0f;     // stats of where(x<0.1, 0, x)
    float mnt = (mn >= 0.1f) ? mn : 0.0f;
    idx_out[plane]   = mi;
    scale_out[plane] = 1.0f / (mxt - mnt + 1e-6f);
    mn_out[plane]    = mnt;
  }
}

// ---------------- Pass 2: preds from argmax ---------------------------------
__global__ __launch_bounds__(256) void preds_kernel(
    const float* __restrict__ hm, const int* __restrict__ idx,
    float* __restrict__ preds) {
  int p = blockIdx.x * blockDim.x + threadIdx.x;
  if (p >= NPLANES) return;
  const float* plane = hm + (size_t)p * PLANE;
  int i0 = idx[p];
  int px = i0 & (WW - 1);
  int py = i0 >> 8;                               // idx0 // 256
  auto g = [&](int dy, int dx) {
    int yy = min(max(py + dy, 0), HH - 1);
    int xx = min(max(px + dx, 0), WW - 1);
    return plane[yy * WW + xx];
  };
  float dxv = g(0, 1) - g(0, -1);
  float dyv = g(1, 0) - g(-1, 0);
  bool valid = (px > 0) && (px < 63) && (py > 0) && (py < 63);
  float sx = (float)((dxv > 0.0f) - (dxv < 0.0f));
  float sy = (float)((dyv > 0.0f) - (dyv < 0.0f));
  preds[2 * p + 0] = (float)(px + 1) + (valid ? 0.25f * sx : 0.0f) - 0.5f;
  preds[2 * p + 1] = (float)(py + 1) + (valid ? 0.25f * sy : 0.0f) - 0.5f;
}

// ---------------- Pass 3 helpers (template-unrolled, all params immediate) ---
template <int S>
__device__ __forceinline__ float slot_load(const float* __restrict__ img,
                                           int h, bool hb, int w) {
  constexpr Slot e = g_slots.s[S];
  // uniform (SGPR) offsets; row exprs CSE across slots (7 distinct shifts)
  int offA = e.cA * PLANE + (((h + e.shA) & (HH - 1)) << 8);
  int offB = e.cB * PLANE + (((h + e.shB) & (HH - 1)) << 8);
  int off  = (hb ? offB : offA) + w;          // 1 cndmask + 1 add
  return img[off];
}

template <int S, int E>
__device__ __forceinline__ void load_all(float* __restrict__ val,
                                         const float* __restrict__ img,
                                         int h, bool hb, int w) {
  if constexpr (S < E) {
    val[S] = slot_load<S>(img, h, hb, w);
    load_all<S + 1, E>(val, img, h, hb, w);
  }
}

template <int S>
__device__ __forceinline__ float slot_xform(float v, const float2* s_nm, bool hb) {
  constexpr Slot e = g_slots.s[S];
  float2 nm;
  if constexpr (e.cA == e.cB) nm = s_nm[e.cA];
  else { int ch = hb ? e.cB : e.cA; nm = s_nm[ch]; }
  v = (v < 0.1f) ? 0.0f : v;                  // threshold
  v = (v - nm.x) * nm.y;                      // normalize
  float t = v;
  if constexpr (e.p >= 2) t *= v;
  if constexpr (e.p >= 3) t *= v;
  if constexpr (e.p >= 4) t *= v;
  if constexpr (e.mA != 1.0f || e.mB != 1.0f)
    t *= (hb ? e.mB : e.mA);
  return t;
}

template <int R, int E>
__device__ __forceinline__ void rounds(v8f& acc, float& xs_sub,
                                       const float* __restrict__ val,
                                       const float2* s_nm, bool hb, v2f ones) {
  if constexpr (R < E) {
    v2f b;
    b[0] = slot_xform<2 * R + 0>(val[2 * R + 0], s_nm, hb);
    b[1] = slot_xform<2 * R + 1>(val[2 * R + 1], s_nm, hb);
    // D = ones(16x4) x B(4x16) + C  => every D row = column sums of B
    acc = __builtin_amdgcn_wmma_f32_16x16x4_f32(
        false, ones, false, b, (short)0, acc, false, false);
    if constexpr (R == XS_SPLIT_ROUND) xs_sub = acc[0];
    rounds<R + 1, E>(acc, xs_sub, val, s_nm, hb, ones);
  }
}

// ---------------- Pass 3: fused preprocess + WMMA channel-sum ---------------
// Block = 256 threads = 8 waves; each wave produces 16 output pixels of row h.
__global__ __launch_bounds__(256) void preprocess_kernel(
    const float* __restrict__ hm, const float* __restrict__ scales,
    const float* __restrict__ mns, float* __restrict__ out_x,
    float* __restrict__ out_x_) {
  const int blk      = blockIdx.x;           // 16*256*2 blocks
  const int half_row = blk & 1;
  const int h        = (blk >> 1) & (HH - 1);
  const int n        = blk >> 9;

  __shared__ float2 s_nm[NCH];               // {mn, scale} per channel
  const int tid = threadIdx.x;
  if (tid < NCH) s_nm[tid] = make_float2(mns[n * NCH + tid], scales[n * NCH + tid]);
  __syncthreads();

  const int wave  = tid >> 5;
  const int lane  = tid & 31;
  const bool hb   = (lane & 16) != 0;        // which wave half
  const int w     = half_row * 128 + wave * 16 + (lane & 15);
  const float* img = hm + (size_t)n * NCH * PLANE;

  // batch all 48 loads (fits LOADcnt<=63) so they pipeline, then compute
  float val[NSLOT];
  load_all<0, NSLOT>(val, img, h, hb, w);

  v2f ones; ones[0] = 1.0f; ones[1] = 1.0f;
  v8f acc = {};
  float xs_sub = 0.0f;
  rounds<0, NROUND>(acc, xs_sub, val, s_nm, hb, ones);

  if (!hb) {
    float xs = acc[0];
    if (xs != xs) xs = 0.0f;                 // isnan -> 0
    xs = fminf(fmaxf(xs, 0.0f), 1.0f);
    float xsp = fminf(fmaxf(xs_sub, 0.0f), 1.0f);
    size_t o = (size_t)n * PLANE + (size_t)h * WW + w;
    out_x[o]  = xs;
    out_x_[o] = xsp;
  }
}

// ---------------- launch ----------------------------------------------------
extern "C" void kernel_launch(void* const* d_in, const int* in_sizes, int n_in,
                              void* d_out, int out_size, void* d_ws, size_t ws_size,
                              hipStream_t stream) {
  const float* hm = (const float*)d_in[0];
  float* out = (float*)d_out;

  int*   ws_idx   = (int*)d_ws;
  float* ws_scale = (float*)d_ws + NPLANES;
  float* ws_mn    = (float*)d_ws + 2 * NPLANES;

  float* out_preds = out;                       // 16*98*2
  float* out_x     = out + 2 * NPLANES;         // 16*1*256*256
  float* out_x_    = out_x + (size_t)N_IMG * PLANE;

  plane_reduce_kernel<<<NPLANES, 256, 0, stream>>>(
      (const float4*)hm, ws_idx, ws_scale, ws_mn);
  preds_kernel<<<(NPLANES + 255) / 256, 256, 0, stream>>>(hm, ws_idx, out_preds);
  preprocess_kernel<<<N_IMG * HH * 2, 256, 0, stream>>>(
      hm, ws_scale, ws_mn, out_x, out_x_);
}